// Sage_75625784148122
// MI455X (gfx1250) — compile-verified
//
#include <hip/hip_runtime.h>
#include <hip/hip_bf16.h>
#include <math.h>

// ---------------------------------------------------------------------------
// 2-layer GraphSAGE for MI455X (gfx1250, wave32, WGP + 320KB LDS).
// GEMMs: V_WMMA_F32_16X16X4_F32 (full f32, matches f32 reference) with
//        block-shared weight slabs staged through LDS (ds_load_b64 reads).
// Segment-mean: global f32 atomics resolved at L2 (x=102MB < 192MB L2).
// ---------------------------------------------------------------------------

namespace {

constexpr int kN1 = 40000;
constexpr int kN2 = 10000;
constexpr int kE1 = 640000;
constexpr int kE2 = 160000;

typedef __attribute__((ext_vector_type(2))) float v2f;
typedef __attribute__((ext_vector_type(8))) float v8f;

__global__ void zero_f32(float* __restrict__ p, int n) {
  int i = blockIdx.x * blockDim.x + threadIdx.x;
  if (i < n) p[i] = 0.0f;
}

// One 256-thread block per edge: scatter-add 256 features + count.
__global__ void scatter_edges(const float* __restrict__ feat,
                              const int* __restrict__ src,
                              const int* __restrict__ dst,
                              float* __restrict__ sum,
                              float* __restrict__ cnt) {
  const int e = blockIdx.x;
  const int s = src[e];
  const int d = dst[e];
  const float* fp = feat + (size_t)s * 256;
  float* sp = sum + (size_t)d * 256;
  atomicAdd(&sp[threadIdx.x], fp[threadIdx.x]);
  if (threadIdx.x == 0) atomicAdd(&cnt[d], 1.0f);
}

// out[M x NDIM] = act( (Asum/max(cnt,1)) @ Wl + Aself @ Wr + bias )
//
// Block = 256 threads (8 waves) owns a 128-row x 64-col output slab; each
// wave computes 16 rows as four 16x16 WMMA tiles. Both weight matrices'
// 64-col slabs are staged through LDS per 32-deep K-chunk and shared by all
// 8 waves (LDS stored [col][k], row stride 34 floats: even => 8B-aligned
// ds_load_b64; 34*i mod 64 distinct for 16 lanes => conflict-free).
//
// V_WMMA_F32_16X16X4_F32 fragment layouts (ISA 7.12.2):
//   A (16x4, 2 VGPR): lane L holds row m0+(L&15); lanes<16 carry K={k,k+1},
//                     lanes>=16 carry K={k+2,k+3}.
//   B (4x16, 2 VGPR): lane L holds col n+(L&15); same K split by lane half.
//   C/D (16x16, 8 VGPR): VGPR r -> row r (lanes<16) / row r+8 (lanes>=16).
//
// Out-of-range rows (M not a multiple of 128) are CLAMPED on load and
// guarded on store only -- no early return, so __syncthreads() is safe and
// EXEC is all-ones at every WMMA.
template <int KDIM, int NDIM, bool SIGMOID>
__global__ void fused_wmma_gemm(const float* __restrict__ Asum,
                                const float* __restrict__ cnt,
                                const float* __restrict__ Aself,
                                const float* __restrict__ Wl,
                                const float* __restrict__ Wr,
                                const float* __restrict__ bias,
                                float* __restrict__ out,
                                int M) {
  constexpr int KC = 32;            // K-chunk staged in LDS
  constexpr int LDS_STRIDE = KC + 2;  // 34: even (align) + coprime-ish (banks)
  __shared__ float ldsW[2][64][LDS_STRIDE];

  const int lane = threadIdx.x & 31;
  const int wave = threadIdx.x >> 5;
  constexpr int kColBlocks = NDIM / 64;
  const int brow = blockIdx.x / kColBlocks;
  const int n0 = (blockIdx.x % kColBlocks) * 64;
  const int m0 = brow * 128 + wave * 16;

  const int half = lane >> 4;  // 0 -> K pair {0,1}, 1 -> K pair {2,3}
  const int l15 = lane & 15;
  const int arow = min(m0 + l15, M - 1);            // clamped A row
  const float inv = 1.0f / fmaxf(cnt[arow], 1.0f);  // segment-mean folded in

  const float* aSumRow  = Asum  + (size_t)arow * KDIM + 2 * half;
  const float* aSelfRow = Aself + (size_t)arow * KDIM + 2 * half;

  v8f acc[4];
#pragma unroll
  for (int t = 0; t < 4; ++t) acc[t] = v8f{0, 0, 0, 0, 0, 0, 0, 0};

  for (int kc = 0; kc < KDIM; kc += KC) {
    __syncthreads();
    // Cooperative stage of Wl/Wr slabs: [KC x 64] -> LDS [col][k].
    // 256 threads: col = t&63 (coalesced 64-float rows), 4 k-rows per pass.
    for (int i = threadIdx.x; i < 64 * KC; i += 256) {
      const int col = i & 63;
      const int kr = i >> 6;
      ldsW[0][col][kr] = Wl[(size_t)(kc + kr) * NDIM + n0 + col];
      ldsW[1][col][kr] = Wr[(size_t)(kc + kr) * NDIM + n0 + col];
    }
    __syncthreads();

#pragma unroll
    for (int k2 = 0; k2 < KC; k2 += 4) {
      v2f a1 = *(const v2f*)(aSumRow + kc + k2);
      a1 = a1 * inv;  // mean scaling (row-diagonal, commutes with GEMM)
      const v2f a2 = *(const v2f*)(aSelfRow + kc + k2);
      const int koff = k2 + 2 * half;
#pragma unroll
      for (int t = 0; t < 4; ++t) {
        const int lcol = 16 * t + l15;
        const v2f bl = *(const v2f*)&ldsW[0][lcol][koff];
        acc[t] = __builtin_amdgcn_wmma_f32_16x16x4_f32(
            false, a1, false, bl, (short)0, acc[t], false, false);
        const v2f br = *(const v2f*)&ldsW[1][lcol][koff];
        acc[t] = __builtin_amdgcn_wmma_f32_16x16x4_f32(
            false, a2, false, br, (short)0, acc[t], false, false);
      }
    }
  }

#pragma unroll
  for (int t = 0; t < 4; ++t) {
    const int col = n0 + 16 * t + l15;
    const float b = bias[col];
#pragma unroll
    for (int r = 0; r < 8; ++r) {
      const int row = m0 + r + 8 * half;
      float v = acc[t][r] + b;
      v = SIGMOID ? (1.0f / (1.0f + expf(-v))) : fmaxf(v, 0.0f);
      if (row < M) out[(size_t)row * NDIM + col] = v;  // per-lane store guard
    }
  }
}

}  // namespace

extern "C" void kernel_launch(void* const* d_in, const int* in_sizes, int n_in,
                              void* d_out, int out_size, void* d_ws, size_t ws_size,
                              hipStream_t stream) {
  (void)in_sizes; (void)n_in; (void)out_size; (void)ws_size;

  const float* x   = (const float*)d_in[0];
  const float* W1l = (const float*)d_in[1];
  const float* b1  = (const float*)d_in[2];
  const float* W1r = (const float*)d_in[3];
  const float* W2l = (const float*)d_in[4];
  const float* b2  = (const float*)d_in[5];
  const float* W2r = (const float*)d_in[6];
  const int* src1  = (const int*)d_in[7];
  const int* dst1  = (const int*)d_in[8];
  const int* src2  = (const int*)d_in[9];
  const int* dst2  = (const int*)d_in[10];
  float* out = (float*)d_out;

  // Workspace layout (floats): [sum1 | cnt1 | sum2 | cnt2 | h]  (~92.4 MB)
  float* ws   = (float*)d_ws;
  float* sum1 = ws;                              // N1*256
  float* cnt1 = sum1 + (size_t)kN1 * 256;        // N1
  float* sum2 = cnt1 + kN1;                      // N2*256
  float* cnt2 = sum2 + (size_t)kN2 * 256;        // N2
  float* h    = cnt2 + kN2;                      // N1*256
  const size_t zeroCount =
      (size_t)kN1 * 256 + kN1 + (size_t)kN2 * 256 + kN2;

  // 1) zero segment sums/counts (fresh every call -> deterministic replay)
  zero_f32<<<(int)((zeroCount + 255) / 256), 256, 0, stream>>>(ws, (int)zeroCount);

  // 2) layer-1 scatter: sum1[dst1] += x[src1], cnt1[dst1] += 1
  scatter_edges<<<kE1, 256, 0, stream>>>(x, src1, dst1, sum1, cnt1);

  // 3) h = relu(mean1 @ W1l + x[:N1] @ W1r + b1)
  //    blocks: ceil(40000/128)=313 row-blocks x 4 col-blocks = 1252
  fused_wmma_gemm<256, 256, false>
      <<<313 * 4, 256, 0, stream>>>(sum1, cnt1, x, W1l, W1r, b1, h, kN1);

  // 4) layer-2 scatter: sum2[dst2] += h[src2], cnt2[dst2] += 1
  scatter_edges<<<kE2, 256, 0, stream>>>(h, src2, dst2, sum2, cnt2);

  // 5) out = sigmoid(mean2 @ W2l + h[:N2] @ W2r + b2)
  //    blocks: ceil(10000/128)=79 row-blocks x 2 col-blocks = 158
  fused_wmma_gemm<256, 128, true>
      <<<79 * 2, 256, 0, stream>>>(sum2, cnt2, h, W2l, W2r, b2, out, kN2);
}